// GCN_14671608283779
// MI455X (gfx1250) — compile-verified
//
#include <hip/hip_runtime.h>

typedef __attribute__((ext_vector_type(2))) float v2f;
typedef __attribute__((ext_vector_type(8))) float v8f;

#define F_IN   256
#define HIDDEN 64
#define NCLS   40

// ---------------- degree / normalization ----------------

__global__ void k_init_deg(float* deg, int n) {
  int i = blockIdx.x * blockDim.x + threadIdx.x;
  if (i < n) deg[i] = 1.0f;  // self-loop contributes weight 1.0 to its own column
}

__global__ void k_deg_edges(const int* __restrict__ col, const float* __restrict__ ew,
                            float* deg, int e) {
  int i = blockIdx.x * blockDim.x + threadIdx.x;
  if (i < e) atomicAdd(&deg[col[i]], ew[i]);
}

__global__ void k_dinv(float* d, int n) {
  int i = blockIdx.x * blockDim.x + threadIdx.x;
  if (i < n) {
    float v = d[i];
    d[i] = (v > 0.0f) ? rsqrtf(v) : 0.0f;
  }
}

// ---------------- WMMA GEMM:  D = op(A) @ W (+bias) (relu) ----------------
// A: [M,K] row-major f32, W: [K,NOUT] row-major f32, D: [M,NOUT].
// One wave -> 16 rows x NOUT cols.  Full-precision V_WMMA_F32_16X16X4_F32.
// A frag element (m=lane%16, k = kbase + 2*(lane>>4) + v) ; B frag element
// (n=lane%16, k likewise); C/D element (M = v + 8*(lane>>4), N = lane%16).

template <int K, int NOUT, bool RELU_IN, bool RELU_OUT, bool BIAS>
__global__ __launch_bounds__(64) void k_gemm_wmma(const float* __restrict__ A,
                                                  const float* __restrict__ W,
                                                  const float* __restrict__ bias,
                                                  float* __restrict__ D, int M) {
  constexpr int NT = (NOUT + 15) / 16;
  const int wv   = threadIdx.x >> 5;
  const int lane = threadIdx.x & 31;
  const int half = lane >> 4;
  const int l16  = lane & 15;
  const int m0   = (blockIdx.x * 2 + wv) * 16;
  if (m0 >= M) return;  // whole-wave exit (grid is exact for M=100000 anyway)

  v8f acc[NT] = {};

  const float* __restrict__ arow = A + (size_t)(m0 + l16) * K + 2 * half;

  for (int k = 0; k < K; k += 4) {
    v2f a = *(const v2f*)(arow + k);  // 8B-aligned: K even, offsets even
    if (RELU_IN) { a.x = fmaxf(a.x, 0.0f); a.y = fmaxf(a.y, 0.0f); }
    const int kk = k + 2 * half;
#pragma unroll
    for (int t = 0; t < NT; ++t) {
      const int c = t * 16 + l16;
      v2f b;
      if ((NOUT % 16 == 0) || (c < NOUT)) {
        b.x = W[(size_t)kk * NOUT + c];
        b.y = W[(size_t)(kk + 1) * NOUT + c];
      } else {
        b.x = 0.0f; b.y = 0.0f;
      }
      acc[t] = __builtin_amdgcn_wmma_f32_16x16x4_f32(
          /*neg_a=*/false, a, /*neg_b=*/false, b,
          /*c_mod=*/(short)0, acc[t], /*reuse_a=*/false, /*reuse_b=*/false);
    }
  }

#pragma unroll
  for (int t = 0; t < NT; ++t) {
    const int c = t * 16 + l16;
    if ((NOUT % 16 != 0) && (c >= NOUT)) continue;
    const float bv = BIAS ? bias[c] : 0.0f;
#pragma unroll
    for (int v = 0; v < 8; ++v) {
      const int m = m0 + v + 8 * half;
      float val = acc[t][v] + bv;
      if (RELU_OUT) val = fmaxf(val, 0.0f);
      D[(size_t)m * NOUT + c] = val;
    }
  }
}

// ---------------- conv epilogue pieces ----------------

// h[i][f] = bias[f] + dinv[i]^2 * t[i][f]   (self-loop term + bias seed)
__global__ void k_seed(const float* __restrict__ t, const float* __restrict__ dinv,
                       const float* __restrict__ bias, float* __restrict__ h, int total) {
  int i = blockIdx.x * blockDim.x + threadIdx.x;
  if (i >= total) return;
  const int node = i >> 6;  // HIDDEN == 64
  const int f    = i & 63;
  const float di = dinv[node];
  h[i] = bias[f] + di * di * t[i];
}

// One wave per edge; each lane handles 2 features (float2).
// h[col] += dinv[row]*ew*dinv[col] * t[row]
__global__ void k_scatter(const int* __restrict__ row, const int* __restrict__ col,
                          const float* __restrict__ ew, const float* __restrict__ dinv,
                          const float* __restrict__ t, float* __restrict__ h, int E) {
  int gid  = blockIdx.x * blockDim.x + threadIdx.x;
  int e    = gid >> 5;
  int lane = gid & 31;
  if (e >= E) return;
  const int   r   = row[e];
  const int   c   = col[e];
  const float nrm = dinv[r] * ew[e] * dinv[c];
  const float2 v  = *(const float2*)(t + (size_t)r * HIDDEN + lane * 2);
  float* dst = h + (size_t)c * HIDDEN + lane * 2;
  atomicAdd(dst + 0, nrm * v.x);
  atomicAdd(dst + 1, nrm * v.y);
}

// ---------------- launch ----------------

extern "C" void kernel_launch(void* const* d_in, const int* in_sizes, int n_in,
                              void* d_out, int out_size, void* d_ws, size_t ws_size,
                              hipStream_t stream) {
  const float* x   = (const float*)d_in[0];
  const int*   ei  = (const int*)d_in[1];   // [2, E]: row = ei[0..E), col = ei[E..2E)
  const float* ew  = (const float*)d_in[2];
  const float* W1  = (const float*)d_in[3];
  const float* b1  = (const float*)d_in[4];
  const float* Wc1 = (const float*)d_in[5];
  const float* bc1 = (const float*)d_in[6];
  const float* Wc2 = (const float*)d_in[7];
  const float* bc2 = (const float*)d_in[8];
  const float* W2  = (const float*)d_in[9];
  const float* b2  = (const float*)d_in[10];
  float* out = (float*)d_out;

  const int N = in_sizes[0] / F_IN;   // 100000
  const int E = in_sizes[2];          // 1600000
  const int* erow = ei;
  const int* ecol = ei + E;

  // workspace: dinv[N] | h[N*64] | t[N*64]
  float* dinv = (float*)d_ws;
  float* h    = dinv + N;
  float* t    = h + (size_t)N * HIDDEN;

  const int TB = 256;
  const int gemm_grid = (N + 31) / 32;  // 2 waves/block, 16 rows/wave

  // --- normalization ---
  k_init_deg<<<(N + TB - 1) / TB, TB, 0, stream>>>(dinv, N);
  k_deg_edges<<<(E + TB - 1) / TB, TB, 0, stream>>>(ecol, ew, dinv, E);
  k_dinv<<<(N + TB - 1) / TB, TB, 0, stream>>>(dinv, N);

  // --- h = relu(x @ W1 + b1) ---
  k_gemm_wmma<F_IN, HIDDEN, false, true, true>
      <<<gemm_grid, 64, 0, stream>>>(x, W1, b1, h, N);

  // --- conv1: t = h @ Wc1 ; h = bc1 + dinv^2*t ; scatter ; (relu fused into next A-load) ---
  k_gemm_wmma<HIDDEN, HIDDEN, false, false, false>
      <<<gemm_grid, 64, 0, stream>>>(h, Wc1, nullptr, t, N);
  k_seed<<<((N * HIDDEN) + TB - 1) / TB, TB, 0, stream>>>(t, dinv, bc1, h, N * HIDDEN);
  k_scatter<<<((size_t)E * 32 + TB - 1) / TB, TB, 0, stream>>>(erow, ecol, ew, dinv, t, h, E);

  // --- conv2: t = relu(h) @ Wc2 ; h = bc2 + dinv^2*t ; scatter ---
  k_gemm_wmma<HIDDEN, HIDDEN, true, false, false>
      <<<gemm_grid, 64, 0, stream>>>(h, Wc2, nullptr, t, N);
  k_seed<<<((N * HIDDEN) + TB - 1) / TB, TB, 0, stream>>>(t, dinv, bc2, h, N * HIDDEN);
  k_scatter<<<((size_t)E * 32 + TB - 1) / TB, TB, 0, stream>>>(erow, ecol, ew, dinv, t, h, E);

  // --- out = relu(h) @ W2 + b2 ---
  k_gemm_wmma<HIDDEN, NCLS, true, false, true>
      <<<gemm_grid, 64, 0, stream>>>(h, W2, b2, out, N);
}